// GCNNet_32839319945298
// MI455X (gfx1250) — compile-verified
//
#include <hip/hip_runtime.h>

#define N_NODES 50000
#define N_EDGES 800000
#define IN_FEAT 128
#define HIDDEN 256
#define NUM_CLASSES 10
#define NUM_GRAPHS 128
#define BN_EPS 1e-5f

typedef float v2f __attribute__((ext_vector_type(2)));
typedef float v8f __attribute__((ext_vector_type(8)));

// ---------------- utility kernels ----------------

__global__ __launch_bounds__(256) void k_zero(float* __restrict__ p, long n) {
    long i = (long)blockIdx.x * blockDim.x + threadIdx.x;
    long stride = (long)gridDim.x * blockDim.x;
    for (; i < n; i += stride) p[i] = 0.0f;
}

__global__ __launch_bounds__(256) void k_set_one(float* __restrict__ p, int n) {
    int i = blockIdx.x * blockDim.x + threadIdx.x;
    if (i < n) p[i] = 1.0f;   // self-loop contributes 1 to every degree
}

__global__ __launch_bounds__(256) void k_deg_accum(const int* __restrict__ dst,
                                                   float* __restrict__ deg) {
    int e = blockIdx.x * blockDim.x + threadIdx.x;
    if (e < N_EDGES) atomicAdd(&deg[dst[e]], 1.0f);
}

__global__ __launch_bounds__(256) void k_rsqrt_inplace(float* __restrict__ p, int n) {
    int i = blockIdx.x * blockDim.x + threadIdx.x;
    if (i < n) p[i] = rsqrtf(p[i]);
}

// ---------------- fp32 WMMA GEMM: C[M x 256] = A[M x K] @ W[K x 256] ------------
// grid.x = M/16 (row strips), blockDim = 256 (8 waves).
// Wave w computes the two 16x16 tiles at columns w*16 and (w+8)*16.
// V_WMMA_F32_16X16X4_F32 operand layout (ISA 7.12.2):
//   A 16x4 : lanes 0-15 rows M0..M15 hold K={0,1} in v[0],v[1]; lanes 16-31 hold K={2,3}
//   B 4x16 : lanes 0-15 cols N0..N15 hold K={0,1};               lanes 16-31 hold K={2,3}
//   C 16x16: VGPR j -> (row j, col lane) for lanes 0-15; (row j+8, col lane-16) for 16-31
__global__ __launch_bounds__(256) void k_gemm_wmma(const float* __restrict__ A,
                                                   const float* __restrict__ W,
                                                   float* __restrict__ C,
                                                   int K) {
    const int wave = threadIdx.x >> 5;
    const int lane = threadIdx.x & 31;
    const int lo   = lane & 15;
    const int hi   = lane >> 4;          // 0 or 1: selects K pair {0,1} vs {2,3}
    const int m0   = blockIdx.x * 16;
    const int n0a  = wave * 16;
    const int n0b  = (wave + 8) * 16;

    v8f acc0 = {};
    v8f acc1 = {};

    const float* arow = A + (long)(m0 + lo) * K + 2 * hi;
    for (int k = 0; k < K; k += 4) {
        v2f a;
        a.x = arow[k];
        a.y = arow[k + 1];
        const float* w0 = W + (long)(k + 2 * hi) * HIDDEN;
        const float* w1 = w0 + HIDDEN;
        v2f b0, b1;
        b0.x = w0[n0a + lo]; b0.y = w1[n0a + lo];
        b1.x = w0[n0b + lo]; b1.y = w1[n0b + lo];
        acc0 = __builtin_amdgcn_wmma_f32_16x16x4_f32(false, a, false, b0,
                                                     (short)0, acc0, false, false);
        acc1 = __builtin_amdgcn_wmma_f32_16x16x4_f32(false, a, false, b1,
                                                     (short)0, acc1, false, false);
    }

    #pragma unroll
    for (int j = 0; j < 8; ++j) {
        const long row = m0 + j + 8 * hi;
        C[row * HIDDEN + n0a + lo] = acc0[j];
        C[row * HIDDEN + n0b + lo] = acc1[j];
    }
}

// ---------------- edge scatter: out[dst] += norm * tmp[src] --------------------
// One wave per edge; 32 lanes sweep the 256 features in 8 coalesced steps.
__global__ __launch_bounds__(256) void k_edge_agg(const int* __restrict__ srcIdx,
                                                  const int* __restrict__ dstIdx,
                                                  const float* __restrict__ dinv,
                                                  const float* __restrict__ tmp,
                                                  float* __restrict__ out) {
    const int wave = threadIdx.x >> 5;
    const int lane = threadIdx.x & 31;
    const long e = (long)blockIdx.x * 8 + wave;
    if (e >= N_EDGES) return;
    const int s = srcIdx[e];
    const int d = dstIdx[e];
    const float nrm = dinv[s] * dinv[d];
    const float* trow = tmp + (long)s * HIDDEN + lane;
    float* orow = out + (long)d * HIDDEN + lane;
    #pragma unroll
    for (int i = 0; i < 8; ++i)
        atomicAdd(orow + i * 32, trow[i * 32] * nrm);
}

// ---------------- epilogues: self-loop + bias (+BN) + ReLU ---------------------
__global__ __launch_bounds__(256) void k_post_layer1(float* __restrict__ h,
                                                     const float* __restrict__ tmp,
                                                     const float* __restrict__ dinv,
                                                     const float* __restrict__ b,
                                                     const float* __restrict__ gamma,
                                                     const float* __restrict__ beta,
                                                     const float* __restrict__ mu,
                                                     const float* __restrict__ var) {
    const int n = blockIdx.x;
    const int f = threadIdx.x;
    const long idx = (long)n * HIDDEN + f;
    const float dv = dinv[n];
    float v = h[idx] + dv * dv * tmp[idx] + b[f];
    v = (v - mu[f]) * rsqrtf(var[f] + BN_EPS) * gamma[f] + beta[f];
    h[idx] = fmaxf(v, 0.0f);
}

__global__ __launch_bounds__(256) void k_post_hidden(float* __restrict__ h,
                                                     const float* __restrict__ tmp,
                                                     const float* __restrict__ dinv,
                                                     const float* __restrict__ b) {
    const int n = blockIdx.x;
    const int f = threadIdx.x;
    const long idx = (long)n * HIDDEN + f;
    const float dv = dinv[n];
    float v = h[idx] + dv * dv * tmp[idx] + b[f];
    h[idx] = fmaxf(v, 0.0f);
}

// ---------------- global mean pool ---------------------------------------------
__global__ __launch_bounds__(256) void k_pool(const int* __restrict__ batch,
                                              const float* __restrict__ h,
                                              float* __restrict__ pooled,
                                              float* __restrict__ counts) {
    const int wave = threadIdx.x >> 5;
    const int lane = threadIdx.x & 31;
    const int n = blockIdx.x * 8 + wave;
    if (n >= N_NODES) return;
    const int g = batch[n];
    const float* hrow = h + (long)n * HIDDEN + lane;
    float* prow = pooled + (long)g * HIDDEN + lane;
    #pragma unroll
    for (int i = 0; i < 8; ++i)
        atomicAdd(prow + i * 32, hrow[i * 32]);
    if (lane == 0) atomicAdd(&counts[g], 1.0f);
}

__global__ __launch_bounds__(256) void k_pool_div(float* __restrict__ pooled,
                                                  const float* __restrict__ counts) {
    const int g = blockIdx.x;
    const int f = threadIdx.x;
    pooled[(long)g * HIDDEN + f] /= fmaxf(counts[g], 1.0f);
}

// ---------------- head: logits + log_softmax ------------------------------------
__global__ __launch_bounds__(32) void k_head(const float* __restrict__ pooled,
                                             const float* __restrict__ W2,
                                             const float* __restrict__ b2,
                                             float* __restrict__ out) {
    __shared__ float sl[NUM_CLASSES];
    __shared__ float s_m, s_lse;
    const int g = blockIdx.x;
    const int lane = threadIdx.x;
    if (lane < NUM_CLASSES) {
        float acc = b2[lane];
        const float* p = pooled + (long)g * HIDDEN;
        for (int k = 0; k < HIDDEN; ++k)
            acc += p[k] * W2[k * NUM_CLASSES + lane];
        sl[lane] = acc;
    }
    __syncthreads();
    if (lane == 0) {
        float m = sl[0];
        for (int i = 1; i < NUM_CLASSES; ++i) m = fmaxf(m, sl[i]);
        float s = 0.0f;
        for (int i = 0; i < NUM_CLASSES; ++i) s += expf(sl[i] - m);
        s_m = m;
        s_lse = logf(s);
    }
    __syncthreads();
    if (lane < NUM_CLASSES)
        out[g * NUM_CLASSES + lane] = sl[lane] - s_m - s_lse;
}

// ---------------- host launch ----------------------------------------------------

extern "C" void kernel_launch(void* const* d_in, const int* in_sizes, int n_in,
                              void* d_out, int out_size, void* d_ws, size_t ws_size,
                              hipStream_t stream) {
    const float* x      = (const float*)d_in[0];
    const int*   src    = (const int*)d_in[1];              // edge_index row 0
    const int*   dst    = src + N_EDGES;                    // edge_index row 1
    const int*   batch  = (const int*)d_in[2];
    const float* W1     = (const float*)d_in[3];
    const float* b1     = (const float*)d_in[4];
    const float* gamma  = (const float*)d_in[5];
    const float* beta   = (const float*)d_in[6];
    const float* rmean  = (const float*)d_in[7];
    const float* rvar   = (const float*)d_in[8];
    const float* Ws     = (const float*)d_in[9];            // [3,256,256]
    const float* bs     = (const float*)d_in[10];           // [3,256]
    const float* W2     = (const float*)d_in[11];
    const float* b2     = (const float*)d_in[12];
    float* out = (float*)d_out;

    // workspace carve-up (floats)
    float* dinv   = (float*)d_ws;                           // N_NODES (pad to 50048)
    float* bufA   = dinv + 50048;                           // tmp = h @ W  [N, 256]
    float* bufB   = bufA + (long)N_NODES * HIDDEN;          // aggregated h [N, 256]
    float* pooled = bufB + (long)N_NODES * HIDDEN;          // [128, 256]
    float* counts = pooled + NUM_GRAPHS * HIDDEN;           // [128]

    const long feat_elems = (long)N_NODES * HIDDEN;         // 12.8M floats

    // 1) symmetric GCN norm: deg (with self loop) -> rsqrt
    k_set_one<<<(N_NODES + 255) / 256, 256, 0, stream>>>(dinv, N_NODES);
    k_deg_accum<<<(N_EDGES + 255) / 256, 256, 0, stream>>>(dst, dinv);
    k_rsqrt_inplace<<<(N_NODES + 255) / 256, 256, 0, stream>>>(dinv, N_NODES);

    // 2) layer 1: x @ W1 -> scatter -> self-loop + bias + BN + ReLU
    k_gemm_wmma<<<N_NODES / 16, 256, 0, stream>>>(x, W1, bufA, IN_FEAT);
    k_zero<<<2048, 256, 0, stream>>>(bufB, feat_elems);
    k_edge_agg<<<N_EDGES / 8, 256, 0, stream>>>(src, dst, dinv, bufA, bufB);
    k_post_layer1<<<N_NODES, 256, 0, stream>>>(bufB, bufA, dinv, b1,
                                               gamma, beta, rmean, rvar);

    // 3) hidden layers
    for (int i = 0; i < 3; ++i) {
        const float* Wi = Ws + (long)i * HIDDEN * HIDDEN;
        const float* bi = bs + (long)i * HIDDEN;
        k_gemm_wmma<<<N_NODES / 16, 256, 0, stream>>>(bufB, Wi, bufA, HIDDEN);
        k_zero<<<2048, 256, 0, stream>>>(bufB, feat_elems);
        k_edge_agg<<<N_EDGES / 8, 256, 0, stream>>>(src, dst, dinv, bufA, bufB);
        k_post_hidden<<<N_NODES, 256, 0, stream>>>(bufB, bufA, dinv, bi);
    }

    // 4) global mean pool
    k_zero<<<64, 256, 0, stream>>>(pooled, (long)NUM_GRAPHS * HIDDEN + NUM_GRAPHS);
    k_pool<<<N_NODES / 8, 256, 0, stream>>>(batch, bufB, pooled, counts);
    k_pool_div<<<NUM_GRAPHS, 256, 0, stream>>>(pooled, counts);

    // 5) head + log_softmax
    k_head<<<NUM_GRAPHS, 32, 0, stream>>>(pooled, W2, b2, out);
}